// KalmanFilter_2534030704847
// MI455X (gfx1250) — compile-verified
//
#include <hip/hip_runtime.h>

// Kalman filter scan on gfx1250: one workgroup per batch chain, state in LDS,
// all GEMMs via V_WMMA_F32_16X16X4_F32 (fp32 matrix path, wave32 tiles).
// v2: single-wave shuffle-pivot Cholesky (no barriers), inverse-diag solves,
//     fused substitution + mean update, preloaded P tiles (no serialized RMW).

typedef __attribute__((ext_vector_type(2))) float v2f;
typedef __attribute__((ext_vector_type(8))) float v8f;

#define NB 256   // batch
#define NT 200   // time steps
#define NM 32    // measurement dim
#define NK 64    // state dim
#define LP 65    // padded LDS stride for 64-wide matrices (odd -> conflict-free)
#define LW 65    // padded stride for W / HP (32x64)
#define LS 33    // padded stride for S (32x32)

__device__ __forceinline__ v8f wmma4(v2f a, v2f b, v8f c) {
    // D = A(16x4,f32) * B(4x16,f32) + C(16x16,f32)
    return __builtin_amdgcn_wmma_f32_16x16x4_f32(false, a, false, b, (short)0, c,
                                                 false, false);
}

__global__ __launch_bounds__(256) void kalman_scan_kernel(
    const float* __restrict__ g_in,   // [B,T,M]
    const float* __restrict__ g_F,    // [K,K]
    const float* __restrict__ g_Q,    // [K,K]
    const float* __restrict__ g_H,    // [M,K]
    const float* __restrict__ g_R,    // [M,M]
    const float* __restrict__ g_m0,   // [B,K]
    const float* __restrict__ g_P0,   // [B,K,K]
    float* __restrict__ out_mean,     // [B,T,M]
    float* __restrict__ out_cov)      // [B,T,M,M]
{
    __shared__ float sH[NM * LP];     // H
    __shared__ float sP[NK * LP];     // covariance P (persistent across t)
    __shared__ float sT[NK * LP];     // temp T1 = F * P_u
    __shared__ float sW[NM * LW];     // HP, then W = L^-1 * HP (in place)
    __shared__ float sS[NM * LS];     // S, then L (inverse diag stored)
    __shared__ float sm[NK];          // state mean (persistent)
    __shared__ float smu[NK];         // updated mean
    __shared__ float sr[NM];          // residual
    __shared__ float sr2[NM];         // L^-1 * residual (redundant same-value writes)

    const int b    = blockIdx.x;
    const int tid  = threadIdx.x;
    const int lane = tid & 31;
    const int wave = tid >> 5;
    const int fm   = lane & 15;            // fragment row (A) / col (B,C)
    const int koff = (lane >> 4) << 1;     // K sub-offset: 0 or 2
    const int crow = (lane >> 4) << 3;     // C row offset: 0 or 8

    // ---- one-time loads: H (shared), per-batch init state ----
    for (int i = tid; i < NM * NK; i += 256) sH[(i >> 6) * LP + (i & 63)] = g_H[i];
    for (int i = tid; i < NK * NK; i += 256) sP[(i >> 6) * LP + (i & 63)] = g_P0[(size_t)b * NK * NK + i];
    if (tid < NK) sm[tid] = g_m0[b * NK + tid];
    __syncthreads();

    for (int t = 0; t < NT; ++t) {
        // ================= HP = H @ P  -> sW   (8 tiles, 1 per wave) =========
        {
            const int tm = wave >> 2, tn = wave & 3;
            v8f c = {};
            #pragma unroll
            for (int kc = 0; kc < NK; kc += 4) {
                const int k0 = kc + koff;
                v2f a  = { sH[(tm * 16 + fm) * LP + k0], sH[(tm * 16 + fm) * LP + k0 + 1] };
                v2f bb = { sP[k0 * LP + tn * 16 + fm],   sP[(k0 + 1) * LP + tn * 16 + fm] };
                c = wmma4(a, bb, c);
            }
            #pragma unroll
            for (int v = 0; v < 8; ++v)
                sW[(tm * 16 + crow + v) * LW + tn * 16 + fm] = c[v];
        }
        __syncthreads();

        // ========== S = HP @ H^T + R -> sS and out_cov[t]  (4 tiles) =========
        if (wave < 4) {
            const int tm = wave >> 1, tn = wave & 1;
            v8f c;
            #pragma unroll
            for (int v = 0; v < 8; ++v)
                c[v] = g_R[(tm * 16 + crow + v) * NM + tn * 16 + fm];
            #pragma unroll
            for (int kc = 0; kc < NK; kc += 4) {
                const int k0 = kc + koff;
                v2f a  = { sW[(tm * 16 + fm) * LW + k0], sW[(tm * 16 + fm) * LW + k0 + 1] };
                v2f bb = { sH[(tn * 16 + fm) * LP + k0], sH[(tn * 16 + fm) * LP + k0 + 1] }; // B[k][n]=H[n][k]
                c = wmma4(a, bb, c);
            }
            float* oc = out_cov + ((size_t)b * NT + t) * NM * NM;
            #pragma unroll
            for (int v = 0; v < 8; ++v) {
                const int r = tm * 16 + crow + v, col = tn * 16 + fm;
                sS[r * LS + col] = c[v];
                oc[r * NM + col] = c[v];
            }
        }
        __syncthreads();

        // ==== wave0: single-wave Cholesky (shfl pivot broadcast, inv diag) ===
        // ==== wave1: predicted measurement mean + residual (concurrent) ======
        if (wave == 0 && t < NT - 1) {
            const int i = lane;                      // row
            #pragma unroll 1
            for (int j = 0; j < NM; ++j) {
                float s = sS[i * LS + j];
                #pragma unroll 1
                for (int p = 0; p < j; ++p)
                    s -= sS[i * LS + p] * sS[j * LS + p];
                const float sj   = __shfl(s, j);
                const float d    = sqrtf(fmaxf(sj, 1e-20f));
                const float dinv = 1.0f / d;         // uniform across lanes
                if (i == j)      sS[i * LS + j] = dinv;      // store INVERSE diag
                else if (i > j)  sS[i * LS + j] = s * dinv;  // L[i][j]
            }
        } else if (wave == 1) {
            const int mm = lane;
            float hm = 0.f;
            #pragma unroll 8
            for (int k = 0; k < NK; ++k) hm += sH[mm * LP + k] * sm[k];
            out_mean[((size_t)b * NT + t) * NM + mm] = hm;
            if (t < NT - 1)
                sr[mm] = g_in[((size_t)b * NT + t) * NM + mm] - hm;
        }
        __syncthreads();
        if (t == NT - 1) break;

        // == fused: W = L^-1 HP (per-column), redundant r2 solve, mean update ==
        if (tid < NK) {
            const int c = tid;                        // owned W column
            float acc = 0.f;
            #pragma unroll 1
            for (int i = 0; i < NM; ++i) {
                float s = sW[i * LW + c];
                float r = sr[i];
                #pragma unroll 1
                for (int p = 0; p < i; ++p) {
                    const float lp = sS[i * LS + p];  // broadcast read
                    s -= lp * sW[p * LW + c];
                    r -= lp * sr2[p];
                }
                const float dinv = sS[i * LS + i];    // inverse diag
                s *= dinv;
                r *= dinv;
                sW[i * LW + c] = s;
                sr2[i] = r;                           // same-value redundant store
                acc += s * r;                         // (W^T r2) contribution
            }
            smu[c] = sm[c] + acc;                     // m_u
        }
        __syncthreads();

        // ===== mean predict (m = F m_u) + P_u = P - W^T W (16 tiles) =========
        if (tid < NK) {
            float s = 0.f;
            #pragma unroll 8
            for (int k = 0; k < NK; ++k) s += g_F[tid * NK + k] * smu[k];
            sm[tid] = s;
        }
        #pragma unroll
        for (int rep = 0; rep < 2; ++rep) {
            const int tile = wave * 2 + rep;
            const int tm = tile >> 2, tn = tile & 3;
            v8f pold;                                  // preload P tile (batched loads)
            #pragma unroll
            for (int v = 0; v < 8; ++v)
                pold[v] = sP[(tm * 16 + crow + v) * LP + tn * 16 + fm];
            v8f c = {};
            #pragma unroll
            for (int kc = 0; kc < NM; kc += 4) {
                const int k0 = kc + koff;
                v2f a  = { sW[k0 * LW + tm * 16 + fm], sW[(k0 + 1) * LW + tm * 16 + fm] }; // A[m][k]=W[k][m]
                v2f bb = { sW[k0 * LW + tn * 16 + fm], sW[(k0 + 1) * LW + tn * 16 + fm] }; // B[k][n]=W[k][n]
                c = wmma4(a, bb, c);
            }
            #pragma unroll
            for (int v = 0; v < 8; ++v)
                sP[(tm * 16 + crow + v) * LP + tn * 16 + fm] = pold[v] - c[v]; // exactly symmetric
        }
        __syncthreads();

        // ============ T1 = F @ P_u -> sT (16 tiles) ==========================
        #pragma unroll
        for (int rep = 0; rep < 2; ++rep) {
            const int tile = wave * 2 + rep;
            const int tm = tile >> 2, tn = tile & 3;
            v8f c = {};
            #pragma unroll
            for (int kc = 0; kc < NK; kc += 4) {
                const int k0 = kc + koff;
                v2f a  = { g_F[(tm * 16 + fm) * NK + k0], g_F[(tm * 16 + fm) * NK + k0 + 1] };
                v2f bb = { sP[k0 * LP + tn * 16 + fm],    sP[(k0 + 1) * LP + tn * 16 + fm] };
                c = wmma4(a, bb, c);
            }
            #pragma unroll
            for (int v = 0; v < 8; ++v)
                sT[(tm * 16 + crow + v) * LP + tn * 16 + fm] = c[v];
        }
        __syncthreads();

        // ============ P = T1 @ F^T + Q -> sP (16 tiles) ======================
        #pragma unroll
        for (int rep = 0; rep < 2; ++rep) {
            const int tile = wave * 2 + rep;
            const int tm = tile >> 2, tn = tile & 3;
            v8f c;
            #pragma unroll
            for (int v = 0; v < 8; ++v)
                c[v] = g_Q[(tm * 16 + crow + v) * NK + tn * 16 + fm];
            #pragma unroll
            for (int kc = 0; kc < NK; kc += 4) {
                const int k0 = kc + koff;
                v2f a  = { sT[(tm * 16 + fm) * LP + k0],  sT[(tm * 16 + fm) * LP + k0 + 1] };
                v2f bb = { g_F[(tn * 16 + fm) * NK + k0], g_F[(tn * 16 + fm) * NK + k0 + 1] }; // B[k][n]=F[n][k]
                c = wmma4(a, bb, c);
            }
            #pragma unroll
            for (int v = 0; v < 8; ++v)
                sP[(tm * 16 + crow + v) * LP + tn * 16 + fm] = c[v];
        }
        __syncthreads();
    }
}

extern "C" void kernel_launch(void* const* d_in, const int* in_sizes, int n_in,
                              void* d_out, int out_size, void* d_ws, size_t ws_size,
                              hipStream_t stream) {
    (void)in_sizes; (void)n_in; (void)out_size; (void)d_ws; (void)ws_size;
    const float* g_in = (const float*)d_in[0];   // [B,T,M]
    const float* g_F  = (const float*)d_in[1];   // [K,K]
    const float* g_Q  = (const float*)d_in[2];   // [K,K]
    const float* g_H  = (const float*)d_in[3];   // [M,K]
    const float* g_R  = (const float*)d_in[4];   // [M,M]
    const float* g_m0 = (const float*)d_in[5];   // [B,K]
    const float* g_P0 = (const float*)d_in[6];   // [B,K,K]

    float* out_mean = (float*)d_out;                       // [B,T,M]
    float* out_cov  = out_mean + (size_t)NB * NT * NM;     // [B,T,M,M]

    kalman_scan_kernel<<<NB, 256, 0, stream>>>(g_in, g_F, g_Q, g_H, g_R,
                                               g_m0, g_P0, out_mean, out_cov);
}